// CorticalMap_36464272343058
// MI455X (gfx1250) — compile-verified
//
#include <hip/hip_runtime.h>
#include <hip/hip_bf16.h>
#include <math.h>

// ---- reference constants ----
#define SHEET 192
#define L_TOT (SHEET * SHEET)          // 36864
#define AFF_K 21
#define AFF_C 2
#define AFF_LEN (AFF_C * AFF_K * AFF_K) // 882
#define XH 212                          // SHEET + AFF_K - 1
#define LAT_K 25
#define LAT_Q (LAT_K * LAT_K)           // 625
#define EXC 9
#define MAX_RAD 19
#define HOMEO 0.05f
#define STRENGTH 10.0f
#define LAT_ITERS 15
#define PI_F 3.14159265358979f

typedef __attribute__((ext_vector_type(2))) float v2f;
typedef __attribute__((ext_vector_type(8))) float v8f;
typedef __attribute__((ext_vector_type(4))) unsigned int u32x4;
typedef __attribute__((ext_vector_type(4))) int i32x4;
typedef __attribute__((ext_vector_type(8))) int i32x8;

// ---------------------------------------------------------------------------
// init: build envelopes (env 21x21, sre 9x9, lri 25x25 max-normalized),
// zero the output scalar. One block.
// ---------------------------------------------------------------------------
__global__ __launch_bounds__(256) void init_kernel(float* __restrict__ env,
                                                   float* __restrict__ sre,
                                                   float* __restrict__ lri,
                                                   float* __restrict__ out_scalar) {
  // aff_env = cos(d*pi/21)^2 * (d < 10.5); max is 1 at center -> no renorm needed
  for (int idx = threadIdx.x; idx < AFF_K * AFF_K; idx += 256) {
    float dy = (float)(idx / AFF_K) - 10.f, dx = (float)(idx % AFF_K) - 10.f;
    float d = sqrtf(dy * dy + dx * dx);
    float c = cosf(d * (PI_F / (float)AFF_K));
    env[idx] = (d < 10.5f) ? c * c : 0.f;
  }
  // sre = cos(d*pi/9) * (d < 4.5)  (radially symmetric -> flip(sre)==sre)
  for (int idx = threadIdx.x; idx < EXC * EXC; idx += 256) {
    float dy = (float)(idx / EXC) - 4.f, dx = (float)(idx % EXC) - 4.f;
    float d = sqrtf(dy * dy + dx * dx);
    sre[idx] = (d < 4.5f) ? cosf(d * (PI_F / (float)EXC)) : 0.f;
  }
  // lri = cos(d*pi/25) * (1 - inh) * (d < 12.5), inh = cos(d*pi/4.5)^2 * (d<2.25)
  __shared__ float smax[256];
  float lm = 0.f;
  for (int idx = threadIdx.x; idx < LAT_Q; idx += 256) {
    float dy = (float)(idx / LAT_K) - 12.f, dx = (float)(idx % LAT_K) - 12.f;
    float d = sqrtf(dy * dy + dx * dx);
    float base = cosf(d * (PI_F / (float)LAT_K));
    float ci = cosf(d * (PI_F / 4.5f));
    float inh = (d < 2.25f) ? ci * ci : 0.f;
    float v = (d < 12.5f) ? base * (1.f - inh) : 0.f;
    lri[idx] = v;
    lm = fmaxf(lm, v);
  }
  smax[threadIdx.x] = lm;
  __syncthreads();
  for (int s = 128; s > 0; s >>= 1) {
    if (threadIdx.x < s) smax[threadIdx.x] = fmaxf(smax[threadIdx.x], smax[threadIdx.x + s]);
    __syncthreads();
  }
  float mx = smax[0];
  for (int idx = threadIdx.x; idx < LAT_Q; idx += 256) lri[idx] = lri[idx] / mx;
  if (threadIdx.x == 0) *out_scalar = 0.f;
}

// ---------------------------------------------------------------------------
// aff: each wave handles 16 consecutive locations (one sheet row segment).
// raw_aff row-dots computed as diag( Xt(16x882) * Rfs^T ) via
// V_WMMA_F32_16X16X4_F32 over K-chunks of 4 (fp32, full precision).
// Main loop (kk=0..219) is branch-free (EXEC stays all-1s); the k=880/881
// tail is one extra WMMA with per-lane selects. Per-lane incremental
// (k % 441, channel) tracking avoids the per-iteration magic-division chain.
// A (16x4 f32): lane m=lane&15 holds K = 2*(lane>>4)+{0,1} in the 2 regs.
// B (4x16 f32): same lane mapping with N = lane&15 (B = Rfs^T slice).
// D diag: lanes 0..7 -> acc[lane]; lanes 24..31 -> acc[lane-24] (m = lane-16).
// ---------------------------------------------------------------------------
__global__ __launch_bounds__(256) void aff_kernel(const float* __restrict__ x,
                                                  const float* __restrict__ rfs,
                                                  const float* __restrict__ thr,
                                                  const float* __restrict__ env,
                                                  float* __restrict__ raw_aff,
                                                  float* __restrict__ x_tiles,
                                                  float* __restrict__ aff,
                                                  float* __restrict__ lat,
                                                  float* __restrict__ votes) {
  const int wave = (blockIdx.x * 256 + threadIdx.x) >> 5;  // 2304 waves
  const int lane = threadIdx.x & 31;
  const int m = lane & 15;
  const int khalf = (lane >> 4) << 1;  // 0 or 2
  const int l0 = wave * 16;
  const int i = l0 / SHEET;
  const int j = (l0 % SHEET) + m;
  const int l = l0 + m;
  const float* __restrict__ rrow = rfs + (size_t)l * AFF_LEN;
  float* __restrict__ xrow = x_tiles + (size_t)l * AFF_LEN;
  const float* __restrict__ xb = x + j;  // fold column into the base

  v8f acc = {};
  int k0 = khalf;   // this lane's first K of the chunk
  int km = khalf;   // k0 % 441
  int cc = 0;       // k0 / 441 (channel)
  for (int kk = 0; kk < 220; ++kk) {
    int km1 = km + 1, c1 = cc;
    if (km1 == 441) { km1 = 0; c1 = 1; }      // k0,k0+1 may straddle channels
    const int r0 = km / 21, q0 = km - r0 * 21;
    const int r1 = km1 / 21, q1 = km1 - r1 * 21;
    float p0 = xb[cc * (XH * XH) + (i + r0) * XH + q0] * env[km];
    float p1 = xb[c1 * (XH * XH) + (i + r1) * XH + q1] * env[km1];
    float2 rv = *reinterpret_cast<const float2*>(rrow + k0);  // 8B aligned
    v2f a, b;
    a.x = p0; a.y = p1;
    b.x = fmaxf(rv.x, 0.f); b.y = fmaxf(rv.y, 0.f);  // rfs *= rfs > 0
    xrow[k0] = p0;     // x_tiles base is 4B-aligned only -> scalar stores
    xrow[k0 + 1] = p1;
    acc = __builtin_amdgcn_wmma_f32_16x16x4_f32(false, a, false, b, (short)0, acc,
                                                false, false);
    k0 += 4;
    km += 4;
    if (km >= 441) { km -= 441; cc = 1; }
  }
  // tail chunk: K = 880..883; only khalf==0 lanes carry data (k=880,881)
  {
    const bool valid = (khalf == 0);
    // k=880 -> (c=1, r=20, q=19); k=881 -> (c=1, r=20, q=20)
    float p0 = valid ? xb[(XH * XH) + (i + 20) * XH + 19] * env[439] : 0.f;
    float p1 = valid ? xb[(XH * XH) + (i + 20) * XH + 20] * env[440] : 0.f;
    const int kc = valid ? 880 : 0;  // clamp OOB lanes to a safe in-row address
    float2 rv = *reinterpret_cast<const float2*>(rrow + kc);
    v2f a, b;
    a.x = p0; a.y = p1;
    b.x = valid ? fmaxf(rv.x, 0.f) : 0.f;
    b.y = valid ? fmaxf(rv.y, 0.f) : 0.f;
    if (valid) { xrow[880] = p0; xrow[881] = p1; }
    acc = __builtin_amdgcn_wmma_f32_16x16x4_f32(false, a, false, b, (short)0, acc,
                                                false, false);
  }

  // diagonal extraction + threshold/init
  float raw = 0.f;
  int wm = -1;
#pragma unroll
  for (int r = 0; r < 8; ++r) {
    float cr = acc[r];
    if (lane == r) { raw = cr; wm = r; }
    if (lane == r + 24) { raw = cr; wm = r + 8; }
  }
  if (wm >= 0) {
    const int ll = l0 + wm;
    raw_aff[ll] = raw;
    float a0 = fmaxf(raw - thr[ll], 0.f);
    aff[ll] = a0;
    lat[ll] = a0;
    votes[ll] = 0.f;
  }
}

// ---------------------------------------------------------------------------
// latmax: per location, masked 19x19 argmax over padded lat (LDS-tiled);
// if lat(center) > 0 scatter a 1.0 vote at the winner (races write same value).
// Strict '>' with ascending (h,w) scan == jnp.argmax first-max tie-break.
// ---------------------------------------------------------------------------
__global__ __launch_bounds__(256) void latmax_kernel(const float* __restrict__ lat,
                                                     float* __restrict__ votes) {
  __shared__ float sm[34 * 34];
  __shared__ float mk[MAX_RAD * MAX_RAD];
  const int tx = threadIdx.x & 15, ty = threadIdx.x >> 4;
  const int bx = blockIdx.x % 12, by = blockIdx.x / 12;
  for (int idx = threadIdx.x; idx < 34 * 34; idx += 256) {
    int sy = idx / 34, sx = idx % 34;
    int gy = by * 16 - 9 + sy, gx = bx * 16 - 9 + sx;
    sm[idx] = (gy >= 0 && gy < SHEET && gx >= 0 && gx < SHEET) ? lat[gy * SHEET + gx] : 0.f;
  }
  for (int idx = threadIdx.x; idx < MAX_RAD * MAX_RAD; idx += 256) {
    float dh = (float)(idx / MAX_RAD) - 9.f, dw = (float)(idx % MAX_RAD) - 9.f;
    mk[idx] = (sqrtf(dh * dh + dw * dw) < 9.5f) ? 1.f : 0.f;
  }
  __syncthreads();
  float best = -1.f;
  int bh = 0, bw = 0;
  for (int h = 0; h < MAX_RAD; ++h)
#pragma unroll
    for (int w = 0; w < MAX_RAD; ++w) {
      float v = sm[(ty + h) * 34 + tx + w] * mk[h * MAX_RAD + w];
      if (v > best) { best = v; bh = h; bw = w; }
    }
  float c = sm[(ty + 9) * 34 + tx + 9];
  if (c > 0.f && best > 0.f) {
    int y = by * 16 + ty + bh - 9, x = bx * 16 + tx + bw - 9;
    if (y >= 0 && y < SHEET && x >= 0 && x < SHEET) votes[y * SHEET + x] = 1.0f;
  }
}

// lat = tanh(relu(min(votes,1)*aff - thr) * 10); clear votes for next iter
__global__ __launch_bounds__(256) void latupd_kernel(float* __restrict__ votes,
                                                     const float* __restrict__ aff,
                                                     const float* __restrict__ thr,
                                                     float* __restrict__ lat) {
  const int l = blockIdx.x * 256 + threadIdx.x;
  float v = fminf(votes[l], 1.f);
  lat[l] = tanhf(fmaxf(v * aff[l] - thr[l], 0.f) * STRENGTH);
  votes[l] = 0.f;
}

// 9x9 SAME conv with flip(sre) (LDS-tiled stencil)
__global__ __launch_bounds__(256) void conv_kernel(const float* __restrict__ lat,
                                                   const float* __restrict__ sre,
                                                   float* __restrict__ lat_out) {
  __shared__ float sm[24 * 24];
  __shared__ float sk[EXC * EXC];
  const int tx = threadIdx.x & 15, ty = threadIdx.x >> 4;
  const int bx = blockIdx.x % 12, by = blockIdx.x / 12;
  for (int idx = threadIdx.x; idx < 24 * 24; idx += 256) {
    int sy = idx / 24, sx = idx % 24;
    int gy = by * 16 - 4 + sy, gx = bx * 16 - 4 + sx;
    sm[idx] = (gy >= 0 && gy < SHEET && gx >= 0 && gx < SHEET) ? lat[gy * SHEET + gx] : 0.f;
  }
  if (threadIdx.x < EXC * EXC) sk[threadIdx.x] = sre[threadIdx.x];
  __syncthreads();
  float s = 0.f;
#pragma unroll
  for (int u = 0; u < EXC; ++u)
#pragma unroll
    for (int v = 0; v < EXC; ++v)
      s += sm[(ty + u) * 24 + tx + v] * sk[(EXC - 1 - u) * EXC + (EXC - 1 - v)];
  lat_out[(by * 16 + ty) * SHEET + bx * 16 + tx] = s;
}

// ---------------------------------------------------------------------------
// lat_correlations: wave per location; lanes stride over q=0..624 (coalesced
// lat_weights reads); shfl-xor wave reduce; one atomicAdd per block.
// The lri table (625 f32 = 2.5 KB) is staged into LDS by the Tensor Data
// Mover: wave 0 issues one tensor_load_to_lds (TDM ignores EXEC; scalar
// branch keeps other waves from duplicating the DMA), waits TENSORcnt==0,
// then __syncthreads publishes the tile to all 8 waves.
// ---------------------------------------------------------------------------
__global__ __launch_bounds__(256) void corr_kernel(const float* __restrict__ lat_out,
                                                   const float* __restrict__ lat_w,
                                                   const float* __restrict__ lri,
                                                   float* __restrict__ out_scalar) {
  __shared__ float s_lri[LAT_Q];
  __shared__ float s_part[8];
  if ((threadIdx.x >> 5) == 0) {  // wave 0 only
#if __has_builtin(__builtin_amdgcn_tensor_load_to_lds)
    const unsigned long long ga = (unsigned long long)(uintptr_t)lri;
    const unsigned int lds_addr = (unsigned int)(uintptr_t)(void*)s_lri;
    // D# group0: count=1 | lds_addr | global_addr[56:0] | type=2
    u32x4 g0;
    g0.x = 1u;
    g0.y = lds_addr;
    g0.z = (unsigned int)(ga & 0xFFFFFFFFull);
    g0.w = (unsigned int)((ga >> 32) & 0x1FFFFFFull) | (2u << 30);
    // D# group1: data_size=4B; tensor_dim0=625, tensor_dim1=1;
    // tile_dim0=625, tile_dim1=1; tensor_dim0_stride=625
    i32x8 g1 = { 0x20000,                         // data_size=2 (4 bytes)
                 (int)(625u << 16),               // tensor_dim0[15:0] @ [63:48]
                 (int)(1u << 16),                 // dim0 hi=0, tensor_dim1 lo @ [95:80]
                 (int)(625u << 16),               // dim1 hi=0, tile_dim0 @ [127:112]
                 1,                               // tile_dim1=1, tile_dim2=0
                 625,                             // tensor_dim0_stride[31:0]
                 0, 0 };
    i32x4 z4 = { 0, 0, 0, 0 };
#if __clang_major__ >= 23
    i32x8 z8 = { 0, 0, 0, 0, 0, 0, 0, 0 };
    __builtin_amdgcn_tensor_load_to_lds(g0, g1, z4, z4, z8, 0);
#else
    __builtin_amdgcn_tensor_load_to_lds(g0, g1, z4, z4, 0);
#endif
    __builtin_amdgcn_s_wait_tensorcnt(0);
#else
    for (int idx = threadIdx.x; idx < LAT_Q; idx += 32) s_lri[idx] = lri[idx];
#endif
  }
  __syncthreads();
  const int wave = threadIdx.x >> 5, lane = threadIdx.x & 31;
  const int l = blockIdx.x * 8 + wave;
  const int i = l / SHEET, j = l % SHEET;
  const float* __restrict__ wrow = lat_w + (size_t)l * LAT_Q;
  float s = 0.f;
  for (int q = lane; q < LAT_Q; q += 32) {
    int qh = q / LAT_K, qw = q % LAT_K;
    int yy = i + qh * 2 - 24, xx = j + qw * 2 - 24;  // padc = 12*2, dilation 2
    float lp = (yy >= 0 && yy < SHEET && xx >= 0 && xx < SHEET)
                   ? lat_out[yy * SHEET + xx] : HOMEO;
    s += lp * s_lri[q] * fmaxf(wrow[q], 0.f);
  }
  s *= lat_out[l];
#pragma unroll
  for (int off = 16; off > 0; off >>= 1) s += __shfl_xor(s, off, 32);
  if (lane == 0) s_part[wave] = s;
  __syncthreads();
  if (threadIdx.x == 0) {
    float t = 0.f;
#pragma unroll
    for (int wv = 0; wv < 8; ++wv) t += s_part[wv];
    atomicAdd(out_scalar, t);
  }
}

// ---------------------------------------------------------------------------
extern "C" void kernel_launch(void* const* d_in, const int* in_sizes, int n_in,
                              void* d_out, int out_size, void* d_ws, size_t ws_size,
                              hipStream_t stream) {
  const float* x     = (const float*)d_in[0];  // [1,2,212,212]
  const float* rfs   = (const float*)d_in[1];  // [L,882,1]
  const float* lat_w = (const float*)d_in[2];  // [L,625,1]
  const float* thr   = (const float*)d_in[3];  // [1,1,192,192]

  float* out      = (float*)d_out;
  float* raw_aff  = out;                 // 36864
  float* lat_out  = out + L_TOT;         // 36864
  float* scalar_o = out + 2 * L_TOT;     // 1
  float* x_tiles  = out + 2 * L_TOT + 1; // L*882

  float* ws    = (float*)d_ws;
  float* env   = ws;          // 441  (pad to 512)
  float* sre   = ws + 512;    // 81   (pad to 128)
  float* lri   = ws + 640;    // 625
  float* aff   = ws + 2048;              // L
  float* lat   = ws + 2048 + L_TOT;      // L
  float* votes = ws + 2048 + 2 * L_TOT;  // L

  init_kernel<<<1, 256, 0, stream>>>(env, sre, lri, scalar_o);
  aff_kernel<<<L_TOT / 16 / 8, 256, 0, stream>>>(x, rfs, thr, env, raw_aff, x_tiles,
                                                 aff, lat, votes);
  for (int it = 0; it < LAT_ITERS; ++it) {
    latmax_kernel<<<144, 256, 0, stream>>>(lat, votes);
    latupd_kernel<<<L_TOT / 256, 256, 0, stream>>>(votes, aff, thr, lat);
  }
  conv_kernel<<<144, 256, 0, stream>>>(lat, sre, lat_out);
  corr_kernel<<<L_TOT / 8, 256, 0, stream>>>(lat_out, lat_w, lri, scalar_o);
}